// QuantizedWeight_28406913696351
// MI455X (gfx1250) — compile-verified
//
#include <hip/hip_runtime.h>
#include <stdint.h>

// ---------------------------------------------------------------------------
// AQLM-style dequant: out[p*8 + k] = (cb0[code0[p]][k] + cb1[code1[p]][k]) * s[p] + z[p]
// p = flat (out_group, in_group) pair index, 8192*1024 pairs, 8 floats per pair.
// Memory-bound (~388 MB HBM traffic). Codebooks (4 MB) stay L2-resident.
// ---------------------------------------------------------------------------

typedef float v4f __attribute__((ext_vector_type(4)));
typedef int   v4i __attribute__((ext_vector_type(4)));

#define BLOCK 256
#define PPT   4   // pairs per thread

// The async-to-LDS builtin takes (int4 AS1*, int4 AS3*, imm offset, imm cpol).
// clang rejects direct generic<->AS pointer casts; integer round-trip is
// well-defined here: AS1 == flat value for global memory, and flat LDS
// addresses carry the LDS byte offset in their low 32 bits.
typedef __attribute__((address_space(1))) v4i g_v4i;
typedef __attribute__((address_space(3))) v4i l_v4i;

static __device__ __forceinline__ g_v4i* to_glob(const void* p) {
    return (g_v4i*)(unsigned long long)(uintptr_t)p;
}
static __device__ __forceinline__ l_v4i* to_lds(const void* p) {
    return (l_v4i*)(unsigned int)(uintptr_t)p;
}

#if __has_builtin(__builtin_amdgcn_global_load_async_to_lds_b128)
#define HAVE_ASYNC_LDS 1
#else
#define HAVE_ASYNC_LDS 0
#endif

static __device__ __forceinline__ void wait_async0() {
#if __has_builtin(__builtin_amdgcn_s_wait_asynccnt)
    __builtin_amdgcn_s_wait_asynccnt(0);
#else
    asm volatile("s_wait_asynccnt 0" ::: "memory");
#endif
}

__global__ __launch_bounds__(BLOCK)
void QuantizedWeight_28406913696351_kernel(const int*   __restrict__ codes,
                                           const float* __restrict__ codebooks,
                                           const float* __restrict__ scales,
                                           const float* __restrict__ zeros,
                                           float*       __restrict__ out,
                                           int cb1_off, long long num_pairs) {
    const int tid = threadIdx.x;
    const long long p0 = ((long long)blockIdx.x * BLOCK + tid) * PPT;
    if (p0 >= num_pairs) return;

    const float* __restrict__ cb0 = codebooks;
    const float* __restrict__ cb1 = codebooks + cb1_off;

    int   c[2 * PPT];
    float sv[PPT];
    float zv[PPT];

#if HAVE_ASYNC_LDS
    // Stage the streamed operands through LDS with the CDNA5 async copy path.
    // Each lane's async requests land in its own private LDS slots, so a
    // wave-level s_wait_asynccnt(0) is sufficient (no workgroup barrier).
    __shared__ int   lds_codes [BLOCK * 2 * PPT];  // 8 KB
    __shared__ float lds_scales[BLOCK * PPT];      // 4 KB
    __shared__ float lds_zeros [BLOCK * PPT];      // 4 KB

    // codes: 32 B/lane (two b128), scales/zeros: 16 B/lane (one b128 each)
    __builtin_amdgcn_global_load_async_to_lds_b128(
        to_glob(codes + 2 * p0),      to_lds(&lds_codes[tid * 2 * PPT]),     0, 0);
    __builtin_amdgcn_global_load_async_to_lds_b128(
        to_glob(codes + 2 * p0 + 4),  to_lds(&lds_codes[tid * 2 * PPT + 4]), 0, 0);
    __builtin_amdgcn_global_load_async_to_lds_b128(
        to_glob(scales + p0),         to_lds(&lds_scales[tid * PPT]),        0, 0);
    __builtin_amdgcn_global_load_async_to_lds_b128(
        to_glob(zeros + p0),          to_lds(&lds_zeros[tid * PPT]),         0, 0);

    wait_async0();
    __asm__ volatile("" ::: "memory");  // keep LDS reads below the wait

    #pragma unroll
    for (int j = 0; j < 2 * PPT; ++j) c[j] = lds_codes[tid * 2 * PPT + j];
    #pragma unroll
    for (int j = 0; j < PPT; ++j) {
        sv[j] = lds_scales[tid * PPT + j];
        zv[j] = lds_zeros [tid * PPT + j];
    }
#else
    // Fallback: direct streaming (non-temporal) vector loads.
    v4i ca  = __builtin_nontemporal_load((const v4i*)(codes + 2 * p0));
    v4i cbv = __builtin_nontemporal_load((const v4i*)(codes + 2 * p0 + 4));
    c[0] = ca.x;  c[1] = ca.y;  c[2] = ca.z;  c[3] = ca.w;
    c[4] = cbv.x; c[5] = cbv.y; c[6] = cbv.z; c[7] = cbv.w;
    v4f s4 = __builtin_nontemporal_load((const v4f*)(scales + p0));
    v4f z4 = __builtin_nontemporal_load((const v4f*)(zeros + p0));
    sv[0] = s4.x; sv[1] = s4.y; sv[2] = s4.z; sv[3] = s4.w;
    zv[0] = z4.x; zv[1] = z4.y; zv[2] = z4.z; zv[3] = z4.w;
#endif

    // Gather (L2-resident codebooks, default temporal hint) + fused scale/shift,
    // then stream the result out with non-temporal b128 stores.
    #pragma unroll
    for (int j = 0; j < PPT; ++j) {
        const unsigned c0 = (unsigned)c[2 * j + 0];
        const unsigned c1 = (unsigned)c[2 * j + 1];
        const float*   e0 = cb0 + (unsigned long long)c0 * 8u;
        const float*   e1 = cb1 + (unsigned long long)c1 * 8u;

        v4f a0 = *(const v4f*)(e0);
        v4f a1 = *(const v4f*)(e0 + 4);
        v4f b0 = *(const v4f*)(e1);
        v4f b1 = *(const v4f*)(e1 + 4);

        const float s = sv[j];
        const float z = zv[j];
        v4f r0 = (a0 + b0) * s + z;
        v4f r1 = (a1 + b1) * s + z;

        float* o = out + (p0 + j) * 8;
        __builtin_nontemporal_store(r0, (v4f*)(o));
        __builtin_nontemporal_store(r1, (v4f*)(o + 4));
    }
}

extern "C" void kernel_launch(void* const* d_in, const int* in_sizes, int n_in,
                              void* d_out, int out_size, void* d_ws, size_t ws_size,
                              hipStream_t stream) {
    const int*   codes     = (const int*)d_in[0];    // (8192, 1024, 2) int32
    const float* codebooks = (const float*)d_in[1];  // (2, 65536, 1, 8) f32
    const float* scales    = (const float*)d_in[2];  // (8192, 1024, 1, 1) f32
    const float* zeros     = (const float*)d_in[3];  // (8192, 1024, 1, 1) f32
    float*       out       = (float*)d_out;          // (8192, 8192) f32

    const long long num_pairs = (long long)in_sizes[0] / 2;  // 8,388,608
    const int       cb1_off   = in_sizes[1] / 2;             // 65536*8 floats

    const long long threads = (num_pairs + PPT - 1) / PPT;
    const int       blocks  = (int)((threads + BLOCK - 1) / BLOCK);

    QuantizedWeight_28406913696351_kernel<<<blocks, BLOCK, 0, stream>>>(
        codes, codebooks, scales, zeros, out, cb1_off, num_pairs);
}